// SubtokenEmbedding_16174846836814
// MI455X (gfx1250) — compile-verified
//
#include <hip/hip_runtime.h>
#include <hip/hip_bf16.h>

typedef __attribute__((ext_vector_type(2))) float v2f;
typedef __attribute__((ext_vector_type(8))) float v8f;

constexpr int B_ = 4, S_ = 2048, P_ = 16, L_ = 512, D_ = 1024, C_ = 256;
constexpr int BS = B_ * S_;
constexpr float LN_EPS = 1e-5f;

// -----------------------------------------------------------------------------
// Kernel 1: T[p,c,j] = sum_i byte_emb[c,i] * pos_emb[p,i,j]
// 16 GEMMs of (256x512)x(512x512). One wave computes a 16x64 output strip
// (4 accumulators) so the A fragment is reused 4x. f32 WMMA 16x16x4.
// grid = 256 blocks x 256 threads (8 waves) -> 2048 waves = 16p * 16mt * 8nb.
// -----------------------------------------------------------------------------
__global__ __launch_bounds__(256) void table_gemm(
    const float* __restrict__ byte_emb,   // [256, 512]
    const float* __restrict__ pos_emb,    // [16, 512, 512]
    float* __restrict__ T) {              // [16, 256, 512]
  const int wave = threadIdx.x >> 5;
  const int lane = threadIdx.x & 31;
  const int gw   = blockIdx.x * 8 + wave;   // 0..2047
  const int p    = gw >> 7;                 // 0..15
  const int rem  = gw & 127;
  const int mt   = rem >> 3;                // 0..15  (16-row tile)
  const int nb   = rem & 7;                 // 0..7   (64-col block)
  const int lm   = lane & 15;
  const int lh   = lane >> 4;

  const float* A  = byte_emb + (size_t)(mt * 16 + lm) * L_;   // A row for this lane
  const float* Bp = pos_emb + (size_t)p * L_ * L_;
  const int n0 = nb * 64;

  v8f acc[4] = {};
  for (int k0 = 0; k0 < L_; k0 += 4) {
    const int k = k0 + 2 * lh;                     // ISA A-layout: K = 2*half + {0,1}
    v2f a = *(const v2f*)(A + k);                  // 8B-aligned (k even)
    const float* brow = Bp + (size_t)k * L_;
#pragma unroll
    for (int j = 0; j < 4; ++j) {
      const int n = n0 + j * 16 + lm;
      v2f b;
      b.x = brow[n];
      b.y = brow[L_ + n];
      acc[j] = __builtin_amdgcn_wmma_f32_16x16x4_f32(
          false, a, false, b, (short)0, acc[j], false, false);
    }
  }

  float* Tout = T + ((size_t)p * C_ + mt * 16) * L_;
#pragma unroll
  for (int j = 0; j < 4; ++j) {
    const int n = n0 + j * 16 + lm;
#pragma unroll
    for (int v = 0; v < 8; ++v) {
      Tout[(size_t)(v + 8 * lh) * L_ + n] = acc[j][v];   // lanes 0..15 coalesced 64B
    }
  }
}

// -----------------------------------------------------------------------------
// Kernel 2: e[t,:] = sum_p ( ids[t,p] >= 0 ? T[p, ids[t,p], :] : 0 )
// One 128-thread block per token; float4 per thread; T rows hit in L2 (8MB table).
// -----------------------------------------------------------------------------
__global__ __launch_bounds__(128) void gather_sum(
    const int* __restrict__ ids,          // [BS, 16]
    const float* __restrict__ T,          // [16, 256, 512]
    float* __restrict__ e) {              // [BS, 512]
  const int t   = blockIdx.x;
  const int tid = threadIdx.x;            // 0..127 float4 chunks of 512 floats
  const int* idp = ids + (size_t)t * P_;
  float4 acc = make_float4(0.f, 0.f, 0.f, 0.f);
#pragma unroll
  for (int p = 0; p < P_; ++p) {
    const int id = idp[p];
    if (id >= 0) {
      const float4* row = (const float4*)(T + (size_t)(p * C_ + id) * L_);
      float4 r = row[tid];
      acc.x += r.x; acc.y += r.y; acc.z += r.z; acc.w += r.w;
    }
  }
  ((float4*)(e + (size_t)t * L_))[tid] = acc;
}

// -----------------------------------------------------------------------------
// Kernel 3: out[t,d] = LN( sum_l e[t,l]*up_w[d,l] + up_b[d] ) * gamma + beta
// Block = 16 tokens x D=1024. 8 waves, each owns 8 N-tiles (16x128 strip),
// K-loop of f32 WMMA. LN stats via half-wave shuffles + tiny LDS; output
// written straight from accumulators.
// -----------------------------------------------------------------------------
__global__ __launch_bounds__(256) void up_ln(
    const float* __restrict__ e,          // [BS, 512]
    const float* __restrict__ up_w,       // [1024, 512]
    const float* __restrict__ up_b,       // [1024]
    const float* __restrict__ gamma,      // [1024]
    const float* __restrict__ beta,       // [1024]
    float* __restrict__ out) {            // [BS, 1024]
  __shared__ float ps[16][8];
  __shared__ float qs[16][8];
  __shared__ float meanA[16];
  __shared__ float rstdA[16];

  const int wave = threadIdx.x >> 5;
  const int lane = threadIdx.x & 31;
  const int lm   = lane & 15;
  const int lh   = lane >> 4;
  const int t0   = blockIdx.x * 16;

  v8f acc[8] = {};
  const float* Arow = e + (size_t)(t0 + lm) * L_;
  const int nbase = wave * 128 + lm;      // column for tile j is nbase + j*16

  for (int k0 = 0; k0 < L_; k0 += 4) {
    const int k = k0 + 2 * lh;
    v2f a = *(const v2f*)(Arow + k);
#pragma unroll
    for (int j = 0; j < 8; ++j) {
      const int n = nbase + j * 16;
      // B[k][n] = up_w[n][k]; consecutive k -> contiguous float2 load
      v2f b = *(const v2f*)(up_w + (size_t)n * L_ + k);
      acc[j] = __builtin_amdgcn_wmma_f32_16x16x4_f32(
          false, a, false, b, (short)0, acc[j], false, false);
    }
  }

  // Add bias, accumulate per-row partial sum / sumsq (rows m = v + 8*lh).
  float s[8], q[8];
#pragma unroll
  for (int v = 0; v < 8; ++v) { s[v] = 0.f; q[v] = 0.f; }
#pragma unroll
  for (int j = 0; j < 8; ++j) {
    const float bias = up_b[nbase + j * 16];
#pragma unroll
    for (int v = 0; v < 8; ++v) {
      float x = acc[j][v] + bias;
      acc[j][v] = x;
      s[v] += x;
      q[v] += x * x;
    }
  }
  // Reduce across the 16-lane half (xor masks < 16 stay within the half-wave).
#pragma unroll
  for (int msk = 1; msk <= 8; msk <<= 1) {
#pragma unroll
    for (int v = 0; v < 8; ++v) {
      s[v] += __shfl_xor(s[v], msk, 32);
      q[v] += __shfl_xor(q[v], msk, 32);
    }
  }
  if (lm == 0) {
#pragma unroll
    for (int v = 0; v < 8; ++v) {
      ps[8 * lh + v][wave] = s[v];
      qs[8 * lh + v][wave] = q[v];
    }
  }
  __syncthreads();
  if (threadIdx.x < 16) {
    float tot = 0.f, totq = 0.f;
#pragma unroll
    for (int w = 0; w < 8; ++w) { tot += ps[threadIdx.x][w]; totq += qs[threadIdx.x][w]; }
    const float mean = tot * (1.0f / D_);
    const float var  = totq * (1.0f / D_) - mean * mean;
    meanA[threadIdx.x] = mean;
    rstdA[threadIdx.x] = rsqrtf(var + LN_EPS);
  }
  __syncthreads();

#pragma unroll
  for (int j = 0; j < 8; ++j) {
    const int n = nbase + j * 16;
    const float g = gamma[n], bb = beta[n];
#pragma unroll
    for (int v = 0; v < 8; ++v) {
      const int m = v + 8 * lh;
      out[(size_t)(t0 + m) * D_ + n] = (acc[j][v] - meanA[m]) * rstdA[m] * g + bb;
    }
  }
}

// -----------------------------------------------------------------------------
extern "C" void kernel_launch(void* const* d_in, const int* in_sizes, int n_in,
                              void* d_out, int out_size, void* d_ws, size_t ws_size,
                              hipStream_t stream) {
  (void)in_sizes; (void)n_in; (void)out_size; (void)ws_size;
  const int*   ids      = (const int*)d_in[0];
  const float* byte_emb = (const float*)d_in[1];
  const float* pos_emb  = (const float*)d_in[2];
  const float* up_w     = (const float*)d_in[3];
  const float* up_b     = (const float*)d_in[4];
  const float* ln_g     = (const float*)d_in[5];
  const float* ln_b     = (const float*)d_in[6];
  float* out = (float*)d_out;

  float* T = (float*)d_ws;                        // 16*256*512 f32 = 8 MB
  float* e = T + (size_t)P_ * C_ * L_;            // 8192*512 f32 = 16 MB

  table_gemm<<<256, 256, 0, stream>>>(byte_emb, pos_emb, T);
  gather_sum<<<BS, 128, 0, stream>>>(ids, T, e);
  up_ln<<<BS / 16, 256, 0, stream>>>(e, up_w, up_b, ln_g, ln_b, out);
}